// GumbelTopK_5583457485534
// MI455X (gfx1250) — compile-verified
//
#include <hip/hip_runtime.h>

#define NDIM 8192
#define TPB  256
#define KSEL 32
#define EPT  (NDIM / TPB)   // 32 elements per thread
#define EPSV 1e-8f

// ---- CDNA5 Tensor Data Mover availability (device pass only) ----
#ifdef __HIP_DEVICE_COMPILE__
#if __has_builtin(__builtin_amdgcn_tensor_load_to_lds) && \
    __has_builtin(__builtin_amdgcn_tensor_store_from_lds) && \
    __has_builtin(__builtin_amdgcn_s_wait_tensorcnt)
#define HAVE_TDM 1
#endif
#endif

typedef unsigned int u32x4 __attribute__((ext_vector_type(4)));
typedef int          i32x4 __attribute__((ext_vector_type(4)));
typedef int          i32x8 __attribute__((ext_vector_type(8)));

// Monotone mapping float -> uint32 so unsigned compare == float compare.
__device__ __forceinline__ unsigned fkey(float f) {
  unsigned b = __float_as_uint(f);
  return (b & 0x80000000u) ? ~b : (b | 0x80000000u);
}
__device__ __forceinline__ unsigned long long pack_key(float v, int idx) {
  return ((unsigned long long)fkey(v) << 32) | (unsigned)idx;
}
__device__ __forceinline__ unsigned long long umax64(unsigned long long a,
                                                     unsigned long long b) {
  return a > b ? a : b;
}

#ifdef HAVE_TDM
// Build a TDM descriptor for one contiguous row of NDIM fp32 values
// (2D tensor: tensor_dim0 = NDIM, tensor_dim1 = 1; tile = full row).
// Layout per CDNA5 ISA ch.8 (async_tensor): group0 = {flags, lds_addr,
// global_addr[31:0], global_addr[56:32] | type=2<<30}; group1 packs
// data_size / dims / strides.
__device__ __forceinline__ void tdm_make_row_desc(u32x4 &g0, i32x8 &g1,
                                                  unsigned lds_byte,
                                                  unsigned long long gaddr) {
  g0.x = 1u;                                        // count=1, user mode
  g0.y = lds_byte;                                  // lds_addr (bytes)
  g0.z = (unsigned)(gaddr & 0xFFFFFFFFull);         // global_addr[31:0]
  g0.w = (unsigned)((gaddr >> 32) & 0x01FFFFFFull)  // global_addr[56:32]
         | (2u << 30);                              // type = 2 ("image")
  g1[0] = 0x00020000;                               // data_size=2 -> 4 bytes
  g1[1] = (int)(((unsigned)NDIM & 0xFFFFu) << 16);  // tensor_dim0[15:0] @63:48
  g1[2] = (int)((((unsigned)NDIM >> 16) & 0xFFFFu)  // tensor_dim0[31:16]
                | (1u << 16));                      // tensor_dim1 = 1
  g1[3] = (int)(((unsigned)NDIM & 0xFFFFu) << 16);  // tile_dim0 = NDIM @127:112
  g1[4] = 1;                                        // tile_dim1 = 1
  g1[5] = NDIM;                                     // tensor_dim0_stride lo32
  g1[6] = 0;                                        // stride hi / dim1_stride
  g1[7] = 0;
}
#endif

__global__ __launch_bounds__(TPB)
void gumbel_topk_softmax_kernel(const float* __restrict__ logits,
                                const float* __restrict__ u,
                                float* __restrict__ out) {
  __shared__ float P[NDIM];                 // u row -> perturbed -> output row
  __shared__ unsigned long long red[TPB];   // argmax reduction keys
  __shared__ int topIdx[KSEL];

  const int t = threadIdx.x;
  const unsigned long long row = blockIdx.x;
  const float* lrow = logits + row * NDIM;
  const float* urow = u + row * NDIM;
  float* orow = out + row * NDIM;

#ifdef HAVE_TDM
  // Wave 0: DMA the u row into LDS via the Tensor Data Mover.
  if ((t >> 5) == 0) {
    u32x4 g0; i32x8 g1;
    i32x4 z4 = {0, 0, 0, 0};
    i32x8 z8 = {0, 0, 0, 0, 0, 0, 0, 0};
    tdm_make_row_desc(g0, g1, (unsigned)(unsigned long long)(void*)P,
                      (unsigned long long)urow);
    __builtin_amdgcn_tensor_load_to_lds(g0, g1, z4, z4, z8, 0);
    __builtin_amdgcn_s_wait_tensorcnt(0);
  }
  __syncthreads();
#endif

  // ---- Phase 1: perturbed = logit + gumbel(u), in place in LDS; track
  //      per-thread argmax over its strided slots j = t + 256*i (bank-clean).
  float lm = -__builtin_inff();
  int   li = t;
  #pragma unroll
  for (int i = 0; i < EPT; ++i) {
    const int j = t + (i << 8);
#ifdef HAVE_TDM
    const float uu = P[j];
#else
    const float uu = urow[j];
#endif
    const float lg = lrow[j];
    const float g  = -__logf(-__logf(uu + EPSV) + EPSV);
    const float p  = lg + g;
    P[j] = p;
    if (p > lm) { lm = p; li = j; }
  }
  unsigned long long myKey = pack_key(lm, li);
  __syncthreads();

  // ---- Phase 2: K rounds of block argmax; winner invalidates & rescans.
  for (int k = 0; k < KSEL; ++k) {
    red[t] = myKey;
    __syncthreads();
    if (t < 128) red[t] = umax64(red[t], red[t + 128]);
    __syncthreads();
    if (t < 64)  red[t] = umax64(red[t], red[t + 64]);
    __syncthreads();
    if (t < 32) {                      // exactly one wave32: shuffle finish
      unsigned long long v = umax64(red[t], red[t + 32]);
      v = umax64(v, __shfl_down(v, 16, 32));
      v = umax64(v, __shfl_down(v, 8, 32));
      v = umax64(v, __shfl_down(v, 4, 32));
      v = umax64(v, __shfl_down(v, 2, 32));
      v = umax64(v, __shfl_down(v, 1, 32));
      if (t == 0) red[0] = v;
    }
    __syncthreads();
    const int jw = (int)(unsigned)red[0];
    if (t == 0) topIdx[k] = jw;
    if (t == (jw & (TPB - 1))) {       // owning thread: invalidate + rescan
      P[jw] = -__builtin_inff();
      float m = -__builtin_inff();
      int mi = t;
      #pragma unroll
      for (int i = 0; i < EPT; ++i) {
        const int j = t + (i << 8);
        const float v = P[j];
        if (v > m) { m = v; mi = j; }
      }
      myKey = pack_key(m, mi);
    }
    __syncthreads();
  }

  // ---- Phase 3: zero the LDS row, scatter masked softmax of ORIGINAL
  //      logits at the K winners (lanes 0..31 == one wave == K).
  #pragma unroll
  for (int i = 0; i < EPT; ++i) P[t + (i << 8)] = 0.0f;
  __syncthreads();
  if (t < KSEL) {
    const int j = topIdx[t];
    const float lv = lrow[j];          // L2-hot scattered reload
    float m = lv;
    m = fmaxf(m, __shfl_xor(m, 16, 32));
    m = fmaxf(m, __shfl_xor(m, 8, 32));
    m = fmaxf(m, __shfl_xor(m, 4, 32));
    m = fmaxf(m, __shfl_xor(m, 2, 32));
    m = fmaxf(m, __shfl_xor(m, 1, 32));
    const float e = __expf(lv - m);
    float s = e;
    s += __shfl_xor(s, 16, 32);
    s += __shfl_xor(s, 8, 32);
    s += __shfl_xor(s, 4, 32);
    s += __shfl_xor(s, 2, 32);
    s += __shfl_xor(s, 1, 32);
    P[j] = e / s;
  }
  __syncthreads();

#ifdef HAVE_TDM
  // Wave 0: DMA the finished row LDS -> global via the TDM.
  if ((t >> 5) == 0) {
    u32x4 g0; i32x8 g1;
    i32x4 z4 = {0, 0, 0, 0};
    i32x8 z8 = {0, 0, 0, 0, 0, 0, 0, 0};
    tdm_make_row_desc(g0, g1, (unsigned)(unsigned long long)(void*)P,
                      (unsigned long long)orow);
    __builtin_amdgcn_tensor_store_from_lds(g0, g1, z4, z4, z8, 0);
    __builtin_amdgcn_s_wait_tensorcnt(0);
  }
#else
  #pragma unroll
  for (int i = 0; i < EPT; ++i) orow[t + (i << 8)] = P[t + (i << 8)];
#endif
}

extern "C" void kernel_launch(void* const* d_in, const int* in_sizes, int n_in,
                              void* d_out, int out_size, void* d_ws, size_t ws_size,
                              hipStream_t stream) {
  (void)in_sizes; (void)n_in; (void)out_size; (void)d_ws; (void)ws_size;
  const float* logits = (const float*)d_in[0];
  const float* u      = (const float*)d_in[1];
  float* out          = (float*)d_out;
  gumbel_topk_softmax_kernel<<<NDIM, TPB, 0, stream>>>(logits, u, out);
}